// StochasticGAT_47399259079304
// MI455X (gfx1250) — compile-verified
//
#include <hip/hip_runtime.h>
#include <hip/hip_bf16.h>
#include <math.h>

// ---------------------------------------------------------------------------
// GATv2 x2 on gfx1250 (MI455X): fp32 WMMA GEMMs + L2-resident edge phase.
// ---------------------------------------------------------------------------

#define GAT_N   100000
#define GAT_E   800000
#define GAT_FIN 128
#define GAT_H   4
#define GAT_D   64
#define GAT_HD  256          // H*D
#define NEG_SLOPE 0.2f

typedef __attribute__((ext_vector_type(2))) float v2f;
typedef __attribute__((ext_vector_type(8))) float v8f;

// ---------------------------------------------------------------------------
// fp32 atomic max via int ordering trick (no NaNs in this workload).
// Memory must be initialized to -inf.
// ---------------------------------------------------------------------------
__device__ __forceinline__ void atomicMaxFloat(float* addr, float val) {
    if (val >= 0.0f) {
        atomicMax((int*)addr, __float_as_int(val));
    } else {
        atomicMin((unsigned int*)addr, (unsigned int)__float_as_int(val));
    }
}

// ---------------------------------------------------------------------------
// GEMM + bias: out[M,Ncol] = A[M,K] @ W[K,Ncol] + bias, fp32 WMMA 16x16x4.
// One wave computes one 16x16 C tile; 8 waves/block walk 8 M-tiles.
// A-frag: lane L holds A[M = L%16][k0 + (L/16)*2 + {0,1}]   (2 VGPRs)
// B-frag: lane L holds B[k0 + (L/16)*2 + {0,1}][N = L%16]
// C-frag: VGPR r holds row (r + (L/16)*8), col L%16.
// M=100000 and Ncol=256 are multiples of 16; K multiples of 4 -> no K guard.
// ---------------------------------------------------------------------------
__global__ __launch_bounds__(256) void gemm_bias_wmma(
    const float* __restrict__ A, const float* __restrict__ W,
    const float* __restrict__ bias, float* __restrict__ out,
    int M, int K, int Ncol)
{
    const int lane  = threadIdx.x;          // 0..31 (wave32)
    const int wave  = threadIdx.y;          // 0..7
    const int tileM = blockIdx.x * 8 + wave;
    const int numTilesM = M >> 4;
    if (tileM >= numTilesM) return;         // wave-uniform guard
    const int tileN = blockIdx.y;

    const int half = lane >> 4;             // 0: lanes 0-15, 1: lanes 16-31
    const int lm   = lane & 15;
    const int row  = tileM * 16 + lm;       // A row owned by this lane
    const int col  = tileN * 16 + lm;       // B/C column owned by this lane

    const float* __restrict__ arow = A + (size_t)row * K;

    v8f acc = {};
    for (int k0 = 0; k0 < K; k0 += 4) {
        const int ka = k0 + half * 2;
        v2f a = *(const v2f*)(arow + ka);                    // contiguous b64 load
        v2f b;
        b.x = W[(size_t)ka * Ncol + col];
        b.y = W[(size_t)(ka + 1) * Ncol + col];
        // D = A x B + C   (fp32, full reference precision)
        acc = __builtin_amdgcn_wmma_f32_16x16x4_f32(
            /*neg_a=*/false, a, /*neg_b=*/false, b,
            /*c_mod=*/(short)0, acc, /*reuse_a=*/false, /*reuse_b=*/false);
    }

    const float bv = bias[col];
#pragma unroll
    for (int r = 0; r < 8; ++r) {
        const int orow = tileM * 16 + half * 8 + r;
        out[(size_t)orow * Ncol + col] = acc[r] + bv;
    }
}

// ---------------------------------------------------------------------------
// Pass 1: per-edge attention logits + segment max.
// One wave per edge; lane l owns features [8l, 8l+8) (all within head l/8).
// ---------------------------------------------------------------------------
__device__ __forceinline__ float lrelu(float x) {
    return x > 0.0f ? x : NEG_SLOPE * x;
}

__global__ __launch_bounds__(256) void edge_score_kernel(
    const float* __restrict__ fs, const float* __restrict__ fd,
    const float* __restrict__ attn,           // [H*D] flattened
    const int* __restrict__ src, const int* __restrict__ dst,
    float* __restrict__ score,                // [E,H]
    float* __restrict__ mmax)                 // [N,H], pre-set to -inf
{
    const int e    = blockIdx.x * 8 + (threadIdx.x >> 5);
    const int lane = threadIdx.x & 31;
    if (e >= GAT_E) return;

    const int s = src[e];
    const int d = dst[e];
    const size_t base = (size_t)lane * 8;

    const float4* pfs = (const float4*)(fs + (size_t)s * GAT_HD + base);
    const float4* pfd = (const float4*)(fd + (size_t)d * GAT_HD + base);
    const float4* pat = (const float4*)(attn + base);
    float4 a0 = pfs[0], a1 = pfs[1];
    float4 b0 = pfd[0], b1 = pfd[1];
    float4 t0 = pat[0], t1 = pat[1];

    float sum = t0.x * lrelu(a0.x + b0.x) + t0.y * lrelu(a0.y + b0.y)
              + t0.z * lrelu(a0.z + b0.z) + t0.w * lrelu(a0.w + b0.w)
              + t1.x * lrelu(a1.x + b1.x) + t1.y * lrelu(a1.y + b1.y)
              + t1.z * lrelu(a1.z + b1.z) + t1.w * lrelu(a1.w + b1.w);

    // Reduce across the 8 lanes of each head group (lanes 8h..8h+7).
    sum += __shfl_xor(sum, 1, 32);
    sum += __shfl_xor(sum, 2, 32);
    sum += __shfl_xor(sum, 4, 32);

    if ((lane & 7) == 0) {
        const int h = lane >> 3;
        score[(size_t)e * GAT_H + h] = sum;
        atomicMaxFloat(&mmax[(size_t)d * GAT_H + h], sum);
    }
}

// ---------------------------------------------------------------------------
// Pass 2: ex = exp(score - max[dst]); denom[dst] += ex. One thread per (e,h).
// Overwrites score with ex.
// ---------------------------------------------------------------------------
__global__ __launch_bounds__(256) void edge_exp_kernel(
    float* __restrict__ score, const float* __restrict__ mmax,
    const int* __restrict__ dst, float* __restrict__ denom)
{
    const int tid = blockIdx.x * blockDim.x + threadIdx.x;
    if (tid >= GAT_E * GAT_H) return;
    const int e = tid >> 2;
    const int h = tid & 3;
    const int d = dst[e];
    const float ex = __expf(score[tid] - mmax[(size_t)d * GAT_H + h]);
    score[tid] = ex;
    atomicAdd(&denom[(size_t)d * GAT_H + h], ex);
}

// ---------------------------------------------------------------------------
// Pass 3: out[dst] += (ex/denom[dst]) * fs[src]. One wave per edge.
// fs/fd/out are N*HD fp32, 102.4MB each (307MB together).
// ---------------------------------------------------------------------------
__global__ __launch_bounds__(256) void aggregate_kernel(
    const float* __restrict__ fs, const float* __restrict__ ex,
    const float* __restrict__ denom,
    const int* __restrict__ src, const int* __restrict__ dst,
    float* __restrict__ out)                  // [N,H*D], pre-zeroed
{
    const int e    = blockIdx.x * 8 + (threadIdx.x >> 5);
    const int lane = threadIdx.x & 31;
    if (e >= GAT_E) return;

    const int s = src[e];
    const int d = dst[e];
    const int h = lane >> 3;
    const float alpha = ex[(size_t)e * GAT_H + h] / denom[(size_t)d * GAT_H + h];

    const size_t base = (size_t)lane * 8;
    const float4* pfs = (const float4*)(fs + (size_t)s * GAT_HD + base);
    float4 f0 = pfs[0], f1 = pfs[1];
    float* po = out + (size_t)d * GAT_HD + base;
    atomicAdd(po + 0, alpha * f0.x);
    atomicAdd(po + 1, alpha * f0.y);
    atomicAdd(po + 2, alpha * f0.z);
    atomicAdd(po + 3, alpha * f0.w);
    atomicAdd(po + 4, alpha * f1.x);
    atomicAdd(po + 5, alpha * f1.y);
    atomicAdd(po + 6, alpha * f1.z);
    atomicAdd(po + 7, alpha * f1.w);
}

// ---------------------------------------------------------------------------
// Helpers: init / zero / ELU / head-mean.
// ---------------------------------------------------------------------------
__global__ void init_softmax_state(float* __restrict__ mmax,
                                   float* __restrict__ denom, int n)
{
    const int i = blockIdx.x * blockDim.x + threadIdx.x;
    if (i < n) { mmax[i] = -INFINITY; denom[i] = 0.0f; }
}

__global__ void zero_kernel(float* __restrict__ p, int n)
{
    const int i = blockIdx.x * blockDim.x + threadIdx.x;
    if (i < n) p[i] = 0.0f;
}

__global__ void elu_kernel(float* __restrict__ p, int n)
{
    const int i = blockIdx.x * blockDim.x + threadIdx.x;
    if (i < n) {
        const float x = p[i];
        p[i] = x > 0.0f ? x : (__expf(x) - 1.0f);
    }
}

__global__ void head_mean_kernel(const float* __restrict__ agg,
                                 float* __restrict__ out, int n)  // n = N*D
{
    const int i = blockIdx.x * blockDim.x + threadIdx.x;
    if (i >= n) return;
    const int node = i / GAT_D;
    const int dcol = i - node * GAT_D;
    const float* p = agg + (size_t)node * GAT_HD + dcol;
    out[i] = 0.25f * (p[0] + p[GAT_D] + p[2 * GAT_D] + p[3 * GAT_D]);
}

// ---------------------------------------------------------------------------
// Orchestration
// ---------------------------------------------------------------------------
extern "C" void kernel_launch(void* const* d_in, const int* in_sizes, int n_in,
                              void* d_out, int out_size, void* d_ws, size_t ws_size,
                              hipStream_t stream)
{
    const float* h     = (const float*)d_in[0];
    const int*   src   = (const int*)  d_in[1];
    const int*   dst   = (const int*)  d_in[2];
    const float* Ws0   = (const float*)d_in[3];
    const float* bs0   = (const float*)d_in[4];
    const float* Wd0   = (const float*)d_in[5];
    const float* bd0   = (const float*)d_in[6];
    const float* attn0 = (const float*)d_in[7];
    const float* Ws1   = (const float*)d_in[8];
    const float* bs1   = (const float*)d_in[9];
    const float* Wd1   = (const float*)d_in[10];
    const float* bd1   = (const float*)d_in[11];
    const float* attn1 = (const float*)d_in[12];
    float* out = (float*)d_out;

    // Workspace carve-up (floats): fs | fd | agg | score | mmax | denom
    float* ws    = (float*)d_ws;
    float* fs    = ws;                                   // N*HD
    float* fd    = fs + (size_t)GAT_N * GAT_HD;          // N*HD
    float* agg   = fd + (size_t)GAT_N * GAT_HD;          // N*HD
    float* score = agg + (size_t)GAT_N * GAT_HD;         // E*H
    float* mmax  = score + (size_t)GAT_E * GAT_H;        // N*H
    float* denom = mmax + (size_t)GAT_N * GAT_H;         // N*H

    const int numTilesM = GAT_N / 16;                    // 6250
    dim3 gemmBlock(32, 8);
    dim3 gemmGrid((numTilesM + 7) / 8, GAT_HD / 16);     // (782, 16)
    const int edgeBlocks = (GAT_E + 7) / 8;              // 1 wave/edge, 8 waves/block
    const int nh  = GAT_N * GAT_H;
    const int eh  = GAT_E * GAT_H;
    const int nhd = GAT_N * GAT_HD;

    // ---------------- Layer 0 ----------------
    init_softmax_state<<<(nh + 255) / 256, 256, 0, stream>>>(mmax, denom, nh);
    zero_kernel<<<(nhd + 255) / 256, 256, 0, stream>>>(agg, nhd);

    gemm_bias_wmma<<<gemmGrid, gemmBlock, 0, stream>>>(h, Ws0, bs0, fs, GAT_N, GAT_FIN, GAT_HD);
    gemm_bias_wmma<<<gemmGrid, gemmBlock, 0, stream>>>(h, Wd0, bd0, fd, GAT_N, GAT_FIN, GAT_HD);

    edge_score_kernel<<<edgeBlocks, 256, 0, stream>>>(fs, fd, attn0, src, dst, score, mmax);
    edge_exp_kernel<<<(eh + 255) / 256, 256, 0, stream>>>(score, mmax, dst, denom);
    aggregate_kernel<<<edgeBlocks, 256, 0, stream>>>(fs, score, denom, src, dst, agg);

    // ELU in place: agg becomes h1
    elu_kernel<<<(nhd + 255) / 256, 256, 0, stream>>>(agg, nhd);

    // ---------------- Layer 1 ----------------
    gemm_bias_wmma<<<gemmGrid, gemmBlock, 0, stream>>>(agg, Ws1, bs1, fs, GAT_N, GAT_HD, GAT_HD);
    gemm_bias_wmma<<<gemmGrid, gemmBlock, 0, stream>>>(agg, Wd1, bd1, fd, GAT_N, GAT_HD, GAT_HD);

    init_softmax_state<<<(nh + 255) / 256, 256, 0, stream>>>(mmax, denom, nh);
    zero_kernel<<<(nhd + 255) / 256, 256, 0, stream>>>(agg, nhd);  // h1 no longer needed

    edge_score_kernel<<<edgeBlocks, 256, 0, stream>>>(fs, fd, attn1, src, dst, score, mmax);
    edge_exp_kernel<<<(eh + 255) / 256, 256, 0, stream>>>(score, mmax, dst, denom);
    aggregate_kernel<<<edgeBlocks, 256, 0, stream>>>(fs, score, denom, src, dst, agg);

    // mean over heads -> [N, D]
    const int nd = GAT_N * GAT_D;
    head_mean_kernel<<<(nd + 255) / 256, 256, 0, stream>>>(agg, out, nd);
}